// CRF_49538152792966
// MI455X (gfx1250) — compile-verified
//
#include <hip/hip_runtime.h>
#include <math.h>

// ---------------------------------------------------------------------------
// Dense mean-field CRF, 96x96, 2 labels, on MI455X (gfx1250, wave32).
//
// Q0 update collapses to:  t[j] = sum_i Q0_i * K(i,j);  x = du + 2t - S;
// Q0 = sigmoid(x).  K is evaluated on the fly; the -d^2 / feature-distance
// exponents (incl. ln Q0_i) are computed as 16x16 tiles of rank-4 inner
// products with V_WMMA_F32_16X16X4_F32, then fed to v_exp_f32 (base-2,
// constants pre-scaled by log2 e).  Column norms ride in the WMMA C operand.
// ---------------------------------------------------------------------------

typedef __attribute__((ext_vector_type(2))) float v2f;
typedef __attribute__((ext_vector_type(8))) float v8f;

#define HH 96
#define WW 96
#define NPIX (HH * WW)        // 9216
#define NTILES (NPIX / 16)    // 576
#define L2E 1.4426950408889634f
#define LN_EPS -18.420680743952367f   // ln(1e-8)

__device__ __forceinline__ float exp2_fast(float x) {
  return __builtin_amdgcn_exp2f(x);   // v_exp_f32
}

// ws layout (floats): gx gy gcol bx by bf bcol du q lq S T   (12 * NPIX)
__global__ void crf_prep(const float* __restrict__ img,
                         const int* __restrict__ mask,
                         float* __restrict__ ws) {
  int p = blockIdx.x * blockDim.x + threadIdx.x;
  if (p >= NPIX) return;
  float* gx   = ws + 0 * NPIX;
  float* gy   = ws + 1 * NPIX;
  float* gcol = ws + 2 * NPIX;
  float* bx   = ws + 3 * NPIX;
  float* by   = ws + 4 * NPIX;
  float* bf   = ws + 5 * NPIX;
  float* bcol = ws + 6 * NPIX;
  float* du   = ws + 7 * NPIX;
  float* q    = ws + 8 * NPIX;
  float* lq   = ws + 9 * NPIX;

  float x = (float)(p % WW);
  float y = (float)(p / WW);
  float f = img[p];
  int   m = mask[p];
  float ns = x * x + y * y;

  const float sg = sqrtf(2.0f * L2E / 18.0f);    // gaussian spatial scale
  const float sb = sqrtf(2.0f * L2E / 5000.0f);  // bilateral spatial scale
  const float sf = sqrtf(2.0f * L2E / 400.0f);   // bilateral feature scale

  gx[p]   = x * sg;
  gy[p]   = y * sg;
  gcol[p] = -L2E * ns * (1.0f / 18.0f);
  bx[p]   = x * sb;
  by[p]   = y * sb;
  bf[p]   = f * sf;
  bcol[p] = -L2E * (ns * (1.0f / 5000.0f) + f * f * (1.0f / 400.0f));

  // unary: m==0 -> u0=-ln(EPS), u1=0 ; m!=0 -> u0=0, u1=-ln(EPS)
  float d = (m == 0) ? LN_EPS : -LN_EPS;   // du = u1 - u0
  du[p] = d;
  q[p]  = 1.0f / (1.0f + expf(-d));
  float lnq = (d > 0.0f) ? -log1pf(expf(-d)) : (d - log1pf(expf(d)));
  lq[p] = L2E * lnq;
}

// One wave32 per 16-wide j-tile; loop over all i-tiles.
// Exponents per 16x16 tile pair via two V_WMMA_F32_16X16X4_F32:
//   gauss:     A_i=(xi*sg, yi*sg, 1, gcol_i+lq_i)  B_j=(xj*sg, yj*sg, gcol_j, 1)  C=0
//   bilateral: A_i=(xi*sb, yi*sb, fi*sf, bcol_i+lq_i) B_j=(xj*sb, yj*sb, fj*sf, 1) C=bcol_j
// out[j] = sum_i ( 3*2^Dg + 10*2^Db )  (includes 13*q_j diagonal, removed later)
__global__ void __launch_bounds__(256) crf_matvec(const float* __restrict__ ws,
                                                  const float* __restrict__ lqbuf,
                                                  int useLq,
                                                  float* __restrict__ out) {
  const float* gx   = ws + 0 * NPIX;
  const float* gy   = ws + 1 * NPIX;
  const float* gcol = ws + 2 * NPIX;
  const float* bx   = ws + 3 * NPIX;
  const float* by   = ws + 4 * NPIX;
  const float* bf   = ws + 5 * NPIX;
  const float* bcol = ws + 6 * NPIX;

  int wave = threadIdx.x >> 5;
  int lane = threadIdx.x & 31;
  int lo   = lane & 15;
  bool hi  = lane >= 16;
  int jt = blockIdx.x * 8 + wave;    // 72 blocks * 8 waves = 576 tiles
  int j0 = jt * 16;
  int jj = j0 + lo;

  // column-side (B / C) operands: loop invariant
  float cgx = gx[jj], cgy = gy[jj], cgc = gcol[jj];
  float cbx = bx[jj], cby = by[jj], cbf = bf[jj], cbc = bcol[jj];

  // B 4x16 layout: VGPR0 = K rows {0,2}, VGPR1 = K rows {1,3}
  v2f Bg, Bb;
  Bg.x = hi ? cgc  : cgx;   // b1 = xj*sg | b3 = gcol_j
  Bg.y = hi ? 1.0f : cgy;   // b2 = yj*sg | b4 = 1
  Bb.x = hi ? cbf  : cbx;   // b1 = xj*sb | b3 = fj*sf
  Bb.y = hi ? 1.0f : cby;   // b2 = yj*sb | b4 = 1

  v8f Cz = {0.f, 0.f, 0.f, 0.f, 0.f, 0.f, 0.f, 0.f};
  v8f Cb;
#pragma unroll
  for (int k = 0; k < 8; ++k) Cb[k] = cbc;   // column norm broadcast over rows

  float acc = 0.0f;

  for (int it = 0; it < NTILES; ++it) {
    int ii = it * 16 + lo;
    float rgx = gx[ii], rgy = gy[ii], rgc = gcol[ii];
    float rbx = bx[ii], rby = by[ii], rbf = bf[ii], rbc = bcol[ii];
    float rlq = useLq ? lqbuf[ii] : 0.0f;   // ln2(Q0_i) fold-in; S pass: q==1

    // A 16x4 layout (ISA): VGPR0 = K cols {0,2}, VGPR1 = K cols {1,3}
    v2f Ag, Ab;
    Ag.x = hi ? 1.0f        : rgx;   // a1 = xi*sg | a3 = 1
    Ag.y = hi ? (rgc + rlq) : rgy;   // a2 = yi*sg | a4 = gcol_i + lq_i
    Ab.x = hi ? rbf         : rbx;   // a1 = xi*sb | a3 = fi*sf
    Ab.y = hi ? (rbc + rlq) : rby;   // a2 = yi*sb | a4 = bcol_i + lq_i

    v8f Dg = __builtin_amdgcn_wmma_f32_16x16x4_f32(
        false, Ag, false, Bg, (short)0, Cz, false, false);
    v8f Db = __builtin_amdgcn_wmma_f32_16x16x4_f32(
        false, Ab, false, Bb, (short)0, Cb, false, false);

#pragma unroll
    for (int k = 0; k < 8; ++k)
      acc += 3.0f * exp2_fast(Dg[k]) + 10.0f * exp2_fast(Db[k]);
  }

  // D layout: lane m   (<16): rows 0..7  of col m (across the 8 VGPRs)
  //           lane m+16     : rows 8..15 of col m  -> fold halves together
  acc += __shfl_xor(acc, 16, 32);
  if (lane < 16) out[j0 + lane] = acc;
}

__global__ void crf_update(float* __restrict__ ws,
                           float* __restrict__ pred,
                           int last) {
  int p = blockIdx.x * blockDim.x + threadIdx.x;
  if (p >= NPIX) return;
  float* du = ws + 7 * NPIX;
  float* q  = ws + 8 * NPIX;
  float* lq = ws + 9 * NPIX;
  float* S  = ws + 10 * NPIX;
  float* T  = ws + 11 * NPIX;

  float t = T[p] - 13.0f * q[p];   // remove diagonal (K(i,i)=0): diag = 13*q
  float s = S[p] - 13.0f;
  float x = du[p] + 2.0f * t - s;  // a0 - a1

  if (last) {
    pred[p] = (x < 0.0f) ? 1.0f : 0.0f;   // argmax over {Q0, Q1}, ties -> 0
  } else {
    q[p] = 1.0f / (1.0f + expf(-x));
    float lnq = (x > 0.0f) ? -log1pf(expf(-x)) : (x - log1pf(expf(x)));
    lq[p] = L2E * lnq;
  }
}

extern "C" void kernel_launch(void* const* d_in, const int* in_sizes, int n_in,
                              void* d_out, int out_size, void* d_ws, size_t ws_size,
                              hipStream_t stream) {
  const float* img  = (const float*)d_in[0];
  const int*   mask = (const int*)d_in[1];
  float* ws  = (float*)d_ws;          // needs 12 * 9216 * 4 = 432 KB
  float* out = (float*)d_out;
  float* lq = ws + 9 * NPIX;
  float* S  = ws + 10 * NPIX;
  float* T  = ws + 11 * NPIX;

  crf_prep<<<NPIX / 256, 256, 0, stream>>>(img, mask, ws);
  // column sums S (q == 1 pass)
  crf_matvec<<<NTILES / 8, 256, 0, stream>>>(ws, lq, 0, S);
  for (int it = 0; it < 5; ++it) {
    crf_matvec<<<NTILES / 8, 256, 0, stream>>>(ws, lq, 1, T);
    crf_update<<<NPIX / 256, 256, 0, stream>>>(ws, out, it == 4 ? 1 : 0);
  }
}